// ScatterPolygonTF_52647709114538
// MI455X (gfx1250) — compile-verified
//
#include <hip/hip_runtime.h>
#include <math.h>

// Problem constants from the reference
#define BB 32
#define PP 64
#define NN 8192
#define EPS_CASE 1.0e-4f

#define WAVES_PER_BLOCK 8
#define THREADS (WAVES_PER_BLOCK * 32)
#define N_PER_BLOCK (WAVES_PER_BLOCK * 16)

typedef __attribute__((ext_vector_type(2))) float v2f;
typedef __attribute__((ext_vector_type(8))) float v8f;

// out[b, 0, n] = Re sum_p res ; out[b, 1, n] = Im sum_p res
__global__ __launch_bounds__(THREADS) void scatter_polygon_kernel(
    const float* __restrict__ points,  // [B, P, 2]
    const float* __restrict__ phi,     // [N]
    float* __restrict__ out)           // [B, 2, N]
{
    // Points for this block's b, plus a (0,0) pad that the upper lane-half
    // reads for the WMMA A operand (K=2,3 must be zero) -> unconditional
    // ds_load_b64, no exec masking around the WMMAs.
    __shared__ float spts[PP * 2 + 2];

    const int b   = blockIdx.y;
    const int tid = threadIdx.x;
    if (tid < PP * 2)          spts[tid] = points[b * (PP * 2) + tid];
    else if (tid < PP * 2 + 2) spts[tid] = 0.0f;
    __syncthreads();

    const int lane = tid & 31;
    const int wave = tid >> 5;
    const int half = lane >> 4;   // 0: rows v, 1: rows v+8 of each WMMA tile
    const int nloc = lane & 15;   // column of the 16x16 D tile
    const int n    = blockIdx.x * N_PER_BLOCK + wave * 16 + nloc;

    // q(phi) = [cos(phi), sin(phi) - 1]
    float sph, cph;
    __sincosf(phi[n], &sph, &cph);
    const float q0 = cph;
    const float q1 = sph - 1.0f;
    // |q|^2 >= 0 exactly in f32 (sum of squares), so skip fabs; rcp ~1 ulp.
    const float scale = __builtin_amdgcn_rcpf(fmaf(q0, q0, q1 * q1));

    // B operand of V_WMMA_F32_16X16X4_F32 (4x16, K in VGPRs, N across lanes;
    // lanes 16-31 hold K=2,3 which we zero -> effective K=2 GEMM)
    v2f bmat;
    bmat.x = half ? 0.0f : q0;
    bmat.y = half ? 0.0f : q1;

    const v2f* spts2 = (const v2f*)spts;

    // dots[k][v] = points[p] . q  with p = k*16 + half*8 + v, n = lane's column
    float dots[4][8];
#pragma unroll
    for (int k = 0; k < 4; ++k) {
        // lanes 0-15: row m = k*16 + lane ; lanes 16-31: the zero pad
        const int idx = half ? PP : (k * 16 + nloc);
        const v2f amat = spts2[idx];
        v8f c = {};
        v8f d = __builtin_amdgcn_wmma_f32_16x16x4_f32(
            /*neg_a=*/false, amat, /*neg_b=*/false, bmat,
            /*c_mod=*/(short)0, c, /*reuse_a=*/false, /*reuse_b=*/false);
#pragma unroll
        for (int v = 0; v < 8; ++v) dots[k][v] = d[v];
    }

    // Edge loop: edge p uses e_{p-1} = exp(i * dot[p-1]) and e_p; one sincos
    // per edge (prev sincos carried), plus one extra at each run boundary.
    float acc_re = 0.0f, acc_im = 0.0f;
#pragma unroll
    for (int k = 0; k < 4; ++k) {
        const int pstart = k * 16 + half * 8;
        const int pprev  = (pstart + PP - 1) & (PP - 1);
        const v2f pv0 = spts2[pprev];
        float prevx = pv0.x, prevy = pv0.y;
        float prev_dot = fmaf(prevx, q0, prevy * q1);
        float s0, c0;
        __sincosf(prev_dot, &s0, &c0);
#pragma unroll
        for (int v = 0; v < 8; ++v) {
            const int p = pstart + v;
            const v2f pv = spts2[p];
            const float px = pv.x, py = pv.y;
            const float cur_dot = dots[k][v];
            float s1, c1;
            __sincosf(cur_dot, &s1, &c1);

            const float ddq  = cur_dot - prev_dot;        // d . q
            const float dx   = px - prevx;
            const float dy   = py - prevy;
            const float ddqc = fmaf(dy, q0, -dx * q1);    // d . q_cross
            const float t    = scale * ddqc;

            // case1 = t * (e0 - e1) / ddq ; case2 = t * (-i) * e0
            const float inv = __builtin_amdgcn_rcpf(ddq); // masked when unused
            const float r1  = t * (c0 - c1) * inv;
            const float i1  = t * (s0 - s1) * inv;
            const float r2  = t * s0;
            const float i2  = -t * c0;
            const bool  msk = fabsf(ddq) >= EPS_CASE;
            acc_re += msk ? r1 : r2;
            acc_im += msk ? i1 : i2;

            prevx = px; prevy = py; prev_dot = cur_dot; s0 = s1; c0 = c1;
        }
    }

    // Fold the two lane halves (rows v and v+8 of every tile) -> full sum over p
    acc_re += __shfl_xor(acc_re, 16, 32);
    acc_im += __shfl_xor(acc_im, 16, 32);

    // Half 0 stores Re, half 1 stores Im: one coalesced store per lane, no exec mask.
    const float val = half ? acc_im : acc_re;
    out[(b * 2 + half) * NN + n] = val;
}

extern "C" void kernel_launch(void* const* d_in, const int* in_sizes, int n_in,
                              void* d_out, int out_size, void* d_ws, size_t ws_size,
                              hipStream_t stream) {
    (void)in_sizes; (void)n_in; (void)d_ws; (void)ws_size; (void)out_size;
    const float* points = (const float*)d_in[0];  // [32, 64, 2] f32
    const float* phi    = (const float*)d_in[1];  // [8192] f32
    float* out          = (float*)d_out;          // [32, 2, 8192] f32

    dim3 grid(NN / N_PER_BLOCK, BB, 1);           // 64 x 32 blocks
    dim3 block(THREADS, 1, 1);                    // 256 threads = 8 waves (wave32)
    scatter_polygon_kernel<<<grid, block, 0, stream>>>(points, phi, out);
}